// GAT_61125974557697
// MI455X (gfx1250) — compile-verified
//
#include <hip/hip_runtime.h>
#include <hip/hip_bf16.h>
#include <math.h>

#define IN_CH   128
#define HEADS1  4
#define HID1    32
#define OUT_CH  16
#define LRELU_SLOPE 0.2f

typedef __attribute__((ext_vector_type(2))) float v2f;
typedef __attribute__((ext_vector_type(8))) float v8f;

// ---------------------------------------------------------------- utilities

__global__ __launch_bounds__(256) void fill_f32(float* p, float v, long long n) {
    long long i = (long long)blockIdx.x * blockDim.x + threadIdx.x;
    if (i < n) p[i] = v;
}

__global__ __launch_bounds__(256) void fill_i32(int* p, int v, long long n) {
    long long i = (long long)blockIdx.x * blockDim.x + threadIdx.x;
    if (i < n) p[i] = v;
}

__device__ inline void atomicMaxFloat(float* addr, float value) {
    // monotone int-bits trick; works with -inf init
    if (value >= 0.0f)
        atomicMax((int*)addr, __float_as_int(value));
    else
        atomicMin((unsigned int*)addr, __float_as_uint(value));
}

__device__ inline float leaky(float v) {
    return v > 0.0f ? v : LRELU_SLOPE * v;
}

// ------------------------------------------------ WMMA f32 GEMM  C = A @ B
// A: M x K row-major, B: K x N row-major, C: M x N row-major.
// One wave computes one 16x16 tile of C via V_WMMA_F32_16X16X4_F32, K in
// steps of 4. Requires M % 16 == 0, N % 16 == 0, K % 4 == 0.
__global__ __launch_bounds__(256) void gemm_wmma_f32(
    const float* __restrict__ A, const float* __restrict__ B,
    float* __restrict__ C, int M, int N, int K)
{
    int wave = (blockIdx.x * 256 + threadIdx.x) >> 5;
    int lane = threadIdx.x & 31;
    int ntn  = N >> 4;                 // tiles along N
    int tm   = wave / ntn;
    int tn   = wave - tm * ntn;
    if (tm * 16 >= M) return;          // whole wave exits together (EXEC stays full)

    int row0  = tm << 4;
    int col0  = tn << 4;
    int lrow  = lane & 15;             // A: M index / B: N index for this lane
    int khalf = (lane >> 4) << 1;      // 0 for lanes 0-15, 2 for lanes 16-31

    v8f acc = {};
    for (int k0 = 0; k0 < K; k0 += 4) {
        // A lane layout: v0 = A[row0+lrow, k0+khalf], v1 = A[row0+lrow, k0+khalf+1]
        const float* ap = A + (size_t)(row0 + lrow) * K + (k0 + khalf);
        v2f a; a.x = ap[0]; a.y = ap[1];
        // B lane layout: v0 = B[k0+khalf, col0+lrow], v1 = B[k0+khalf+1, col0+lrow]
        const float* bp = B + (size_t)(k0 + khalf) * N + (col0 + lrow);
        v2f b; b.x = bp[0]; b.y = bp[(size_t)N];
        acc = __builtin_amdgcn_wmma_f32_16x16x4_f32(
            /*neg_a=*/false, a, /*neg_b=*/false, b,
            /*c_mod=*/(short)0, acc, /*reuse_a=*/false, /*reuse_b=*/false);
    }

    // C/D layout: VGPR i holds M = i (lanes 0-15) or M = 8+i (lanes 16-31), N = lane%16
    int mbase = row0 + ((lane >> 4) << 3);
    float* cp = C + (size_t)mbase * N + (col0 + lrow);
#pragma unroll
    for (int i = 0; i < 8; ++i) cp[(size_t)i * N] = acc[i];
}

// ------------------------------------------- attention coefficients per node
// as_[n*heads+h] = sum_c h[n,h,c]*att_s[h,c] ; ad_ likewise
__global__ __launch_bounds__(256) void att_coeff(
    const float* __restrict__ h, const float* __restrict__ att_s,
    const float* __restrict__ att_d, float* __restrict__ as_,
    float* __restrict__ ad_, int n, int heads, int hid)
{
    int i = blockIdx.x * blockDim.x + threadIdx.x;
    if (i >= n * heads) return;
    int node = i / heads, hh = i - node * heads;
    const float* hp = h + (size_t)node * heads * hid + (size_t)hh * hid;
    float s = 0.0f, d = 0.0f;
    for (int c = 0; c < hid; ++c) {
        float hv = hp[c];
        s += hv * att_s[hh * hid + c];
        d += hv * att_d[hh * hid + c];
    }
    as_[i] = s;
    ad_[i] = d;
}

// ---------------------------------------------------- softmax pass 1: max
__global__ __launch_bounds__(256) void edge_seg_max(
    const int* __restrict__ srcA, const int* __restrict__ dstA,
    const float* __restrict__ as_, const float* __restrict__ ad_,
    float* __restrict__ m, long long Etot, long long Eorig, int heads)
{
    long long e = (long long)blockIdx.x * blockDim.x + threadIdx.x;
    if (e >= Etot) return;
    int s, d;
    if (e < Eorig) { s = srcA[e]; d = dstA[e]; }
    else           { s = d = (int)(e - Eorig); }      // self-loop
    for (int hh = 0; hh < heads; ++hh) {
        float v = leaky(as_[(size_t)s * heads + hh] + ad_[(size_t)d * heads + hh]);
        atomicMaxFloat(&m[(size_t)d * heads + hh], v);
    }
}

// ---------------------------------------------------- softmax pass 2: denom
__global__ __launch_bounds__(256) void edge_seg_sum(
    const int* __restrict__ srcA, const int* __restrict__ dstA,
    const float* __restrict__ as_, const float* __restrict__ ad_,
    const float* __restrict__ m, float* __restrict__ denom,
    long long Etot, long long Eorig, int heads)
{
    long long e = (long long)blockIdx.x * blockDim.x + threadIdx.x;
    if (e >= Etot) return;
    int s, d;
    if (e < Eorig) { s = srcA[e]; d = dstA[e]; }
    else           { s = d = (int)(e - Eorig); }
    for (int hh = 0; hh < heads; ++hh) {
        float v = leaky(as_[(size_t)s * heads + hh] + ad_[(size_t)d * heads + hh]);
        float w = expf(v - m[(size_t)d * heads + hh]);
        atomicAdd(&denom[(size_t)d * heads + hh], w);
    }
}

// --------------------------------------- softmax pass 3: weighted scatter-add
// one thread per (edge, head); recomputes alpha, scatters hid channels
__global__ __launch_bounds__(256) void edge_scatter(
    const int* __restrict__ srcA, const int* __restrict__ dstA,
    const float* __restrict__ as_, const float* __restrict__ ad_,
    const float* __restrict__ m, const float* __restrict__ denom,
    const float* __restrict__ h, float* __restrict__ out,
    long long Etot, long long Eorig, int heads, int hid)
{
    long long idx = (long long)blockIdx.x * blockDim.x + threadIdx.x;
    if (idx >= Etot * heads) return;
    long long e = idx / heads;
    int hh = (int)(idx - e * heads);
    int s, d;
    if (e < Eorig) { s = srcA[e]; d = dstA[e]; }
    else           { s = d = (int)(e - Eorig); }
    float v = leaky(as_[(size_t)s * heads + hh] + ad_[(size_t)d * heads + hh]);
    float w = expf(v - m[(size_t)d * heads + hh]);
    float alpha = w / (denom[(size_t)d * heads + hh] + 1e-16f);
    const float* hp = h  + (size_t)s * heads * hid + (size_t)hh * hid;
    float*       op = out + (size_t)d * heads * hid + (size_t)hh * hid;
    for (int c = 0; c < hid; ++c)
        atomicAdd(&op[c], hp[c] * alpha);
}

// ------------------------------------------------------------- bias + ELU
__global__ __launch_bounds__(256) void elu_bias(
    float* __restrict__ y, const float* __restrict__ bias, long long n, int width)
{
    long long i = (long long)blockIdx.x * blockDim.x + threadIdx.x;
    if (i >= n) return;
    float v = y[i] + bias[(int)(i % width)];
    y[i] = v > 0.0f ? v : expm1f(v);
}

// ---------------------------------------------------------------- roots
__global__ __launch_bounds__(256) void find_roots(
    const float* __restrict__ x, const int* __restrict__ batch,
    int* __restrict__ roots, int n)
{
    int i = blockIdx.x * blockDim.x + threadIdx.x;
    if (i >= n) return;
    if (x[(size_t)i * IN_CH] == 0.0f)
        atomicMin(&roots[batch[i]], i);
}

__global__ __launch_bounds__(256) void gather_roots(
    const float* __restrict__ out2, const int* __restrict__ roots,
    const float* __restrict__ bias2, float* __restrict__ dout,
    int g, int oc, int n)
{
    int i = blockIdx.x * blockDim.x + threadIdx.x;
    if (i >= g * oc) return;
    int gg = i / oc, c = i - gg * oc;
    int r = roots[gg];
    if (r >= n) r = 0;   // safety clamp (every group has a type-0 node)
    dout[i] = out2[(size_t)r * oc + c] + bias2[c];
}

// ----------------------------------------------------------------- driver

static inline long long cdiv(long long a, long long b) { return (a + b - 1) / b; }

extern "C" void kernel_launch(void* const* d_in, const int* in_sizes, int n_in,
                              void* d_out, int out_size, void* d_ws, size_t ws_size,
                              hipStream_t stream) {
    (void)n_in; (void)ws_size;
    const float* x        = (const float*)d_in[0];
    const int*   ei       = (const int*)  d_in[1];
    const int*   batch    = (const int*)  d_in[2];
    const float* W1       = (const float*)d_in[3];
    const float* att_src1 = (const float*)d_in[4];
    const float* att_dst1 = (const float*)d_in[5];
    const float* bias1    = (const float*)d_in[6];
    const float* W2       = (const float*)d_in[7];
    const float* att_src2 = (const float*)d_in[8];
    const float* att_dst2 = (const float*)d_in[9];
    const float* bias2    = (const float*)d_in[10];
    float*       dout     = (float*)d_out;

    const int       N    = in_sizes[0] / IN_CH;   // 100000
    const long long E    = in_sizes[1] / 2;       // 1600000
    const int       G    = out_size / OUT_CH;     // 100
    const long long Etot = E + N;                 // + self loops
    const int*      srcA = ei;
    const int*      dstA = ei + E;

    // ---- workspace carve-up (floats) ----
    float* ws = (float*)d_ws;
    size_t off = 0;
    float* h1   = ws + off; off += (size_t)N * 128;   // x @ W1
    float* out1 = ws + off; off += (size_t)N * 128;   // layer-1 aggregate (then ELU'd)
    float* as1  = ws + off; off += (size_t)N * HEADS1;
    float* ad1  = ws + off; off += (size_t)N * HEADS1;
    float* m1   = ws + off; off += (size_t)N * HEADS1;
    float* dn1  = ws + off; off += (size_t)N * HEADS1;
    float* h2   = ws + off; off += (size_t)N * OUT_CH; // elu(out1) @ W2
    float* out2 = ws + off; off += (size_t)N * OUT_CH; // layer-2 aggregate
    float* as2  = ws + off; off += (size_t)N;
    float* ad2  = ws + off; off += (size_t)N;
    float* m2   = ws + off; off += (size_t)N;
    float* dn2  = ws + off; off += (size_t)N;
    int*   roots = (int*)(ws + off);

    const int T = 256;

    // ---- init accumulators ----
    fill_f32<<<cdiv((long long)N * HEADS1, T), T, 0, stream>>>(m1,  -INFINITY, (long long)N * HEADS1);
    fill_f32<<<cdiv((long long)N * HEADS1, T), T, 0, stream>>>(dn1, 0.0f,      (long long)N * HEADS1);
    fill_f32<<<cdiv((long long)N * 128,    T), T, 0, stream>>>(out1, 0.0f,     (long long)N * 128);
    fill_f32<<<cdiv((long long)N,          T), T, 0, stream>>>(m2,  -INFINITY, (long long)N);
    fill_f32<<<cdiv((long long)N,          T), T, 0, stream>>>(dn2, 0.0f,      (long long)N);
    fill_f32<<<cdiv((long long)N * OUT_CH, T), T, 0, stream>>>(out2, 0.0f,     (long long)N * OUT_CH);
    fill_i32<<<cdiv((long long)G,          T), T, 0, stream>>>(roots, N,       (long long)G);

    // ---- layer 1: h1 = x @ W1 (WMMA f32) ----
    {
        long long waves = (long long)(N / 16) * (128 / 16);
        gemm_wmma_f32<<<cdiv(waves * 32, T), T, 0, stream>>>(x, W1, h1, N, 128, 128);
    }
    att_coeff<<<cdiv((long long)N * HEADS1, T), T, 0, stream>>>(
        h1, att_src1, att_dst1, as1, ad1, N, HEADS1, HID1);
    edge_seg_max<<<cdiv(Etot, T), T, 0, stream>>>(srcA, dstA, as1, ad1, m1, Etot, E, HEADS1);
    edge_seg_sum<<<cdiv(Etot, T), T, 0, stream>>>(srcA, dstA, as1, ad1, m1, dn1, Etot, E, HEADS1);
    edge_scatter<<<cdiv(Etot * HEADS1, T), T, 0, stream>>>(
        srcA, dstA, as1, ad1, m1, dn1, h1, out1, Etot, E, HEADS1, HID1);
    elu_bias<<<cdiv((long long)N * 128, T), T, 0, stream>>>(out1, bias1, (long long)N * 128, 128);

    // ---- layer 2: h2 = elu(out1) @ W2 (WMMA f32) ----
    {
        long long waves = (long long)(N / 16) * (OUT_CH / 16);
        gemm_wmma_f32<<<cdiv(waves * 32, T), T, 0, stream>>>(out1, W2, h2, N, OUT_CH, 128);
    }
    att_coeff<<<cdiv((long long)N, T), T, 0, stream>>>(
        h2, att_src2, att_dst2, as2, ad2, N, 1, OUT_CH);
    edge_seg_max<<<cdiv(Etot, T), T, 0, stream>>>(srcA, dstA, as2, ad2, m2, Etot, E, 1);
    edge_seg_sum<<<cdiv(Etot, T), T, 0, stream>>>(srcA, dstA, as2, ad2, m2, dn2, Etot, E, 1);
    edge_scatter<<<cdiv(Etot, T), T, 0, stream>>>(
        srcA, dstA, as2, ad2, m2, dn2, h2, out2, Etot, E, 1, OUT_CH);

    // ---- root selection + gather (heads==1 so mean == identity) ----
    find_roots<<<cdiv((long long)N, T), T, 0, stream>>>(x, batch, roots, N);
    gather_roots<<<cdiv((long long)G * OUT_CH, T), T, 0, stream>>>(
        out2, roots, bias2, dout, G, OUT_CH, N);
}